// CrossEntropyWithPerDomainLoss_51728586113630
// MI455X (gfx1250) — compile-verified
//
#include <hip/hip_runtime.h>
#include <hip/hip_bf16.h>
#include <math.h>

// Problem constants (match reference)
#define N_DOMAINS 8
#define BATCH 4
#define SEQ 2048
#define VOCAB 32000
#define ROWS (BATCH * SEQ)          // 8192 token rows
#define VEC4 (VOCAB / 4)            // 8000 float4 per row (8000 % 32 == 0 -> wave-aligned tail)
#define BLOCK 256                   // 8 wave32 per workgroup
#define NCHUNK ((VEC4 + BLOCK - 1) / BLOCK)   // 32 chunks; chunk 31 covers 64 lanes
#define NBUF 4                      // 4-deep async pipeline (issue 3 ahead)
#define CHUNK_BYTES (BLOCK * 16)    // 4 KB per buffer

#define LOG2E 1.4426950408889634f
#define LN2   0.6931471805599453f

__device__ __forceinline__ float ex2(float x) { return __builtin_amdgcn_exp2f(x); } // v_exp_f32
__device__ __forceinline__ float lg2(float x) { return __builtin_amdgcn_logf(x);  } // v_log_f32

// ---------------------------------------------------------------------------
// Kernel 1: one workgroup per token row. Stream 128KB of logits through a
// 4-deep LDS pipeline fed by CDNA5 async global->LDS copies
// (global_load_async_to_lds_b128, ASYNCcnt-tracked, th:TH_LOAD_NT since the
// stream is touched once and exceeds L2), computing a one-pass online softmax.
// Each logit is read from HBM exactly once -> 1.05GB -> ~45us roofline.
// With ~8 resident workgroups/WGP and 3-4 outstanding 4KB chunks each, ~128KB
// is in flight per WGP, covering HBM latency x per-WGP bandwidth share.
// ---------------------------------------------------------------------------
__global__ __launch_bounds__(BLOCK) void ce_row_kernel(
    const float* __restrict__ logits,
    const int*   __restrict__ labels,
    float*       __restrict__ row_loss)
{
    __shared__ float4 buf[NBUF][BLOCK];           // 4 x 4KB ring buffer
    __shared__ float wm[BLOCK / 32], ws_[BLOCK / 32];

    const int row = blockIdx.x;
    const int tid = threadIdx.x;
    const float* rowp = logits + (size_t)row * VOCAB;

    // LDS byte offset of this lane's slot in buffer 0. A generic pointer to
    // __shared__ is {shared_aperture_hi32, lds_offset}; the low 32 bits are
    // the LDS byte address the async-LDS instruction consumes in VDST.
    const unsigned lds_lane = (unsigned)(unsigned long long)(size_t)&buf[0][tid];

    // Thread 0 prefetches the label logit early (LOADcnt path, overlaps stream)
    float xl = 0.0f;
    if (tid == 0) xl = rowp[labels[row]];

    // Prologue: fill the pipeline with chunks 0..NBUF-2 (all lanes in range).
    #pragma unroll
    for (int k = 0; k < NBUF - 1; ++k) {
        unsigned voff = (unsigned)(k * BLOCK + tid) * 16u;
        unsigned dst  = lds_lane + (unsigned)k * CHUNK_BYTES;
        asm volatile("global_load_async_to_lds_b128 %0, %1, %2 th:TH_LOAD_NT"
                     :: "v"(dst), "v"(voff), "s"(rowp) : "memory");
    }

    float m = -INFINITY, s = 0.0f;
    for (int c = 0; c < NCHUNK; ++c) {
        // Issue chunk c+3 into the ring slot we finished reading at iter c-1.
        // Tail lanes clamp the global offset to the last valid float4 so every
        // wave issues every chunk -> ASYNCcnt bookkeeping is uniform and the
        // wait immediates below are exact. (Redundant tail loads hit a hot
        // line; the compute predicate discards them.)
        if (c + (NBUF - 1) < NCHUNK) {
            // Slot being overwritten was ds_load'ed last iteration; make sure
            // that ds_load fully completed before the async write can land.
            asm volatile("s_wait_dscnt 0x0" ::: "memory");
            unsigned nidx = (unsigned)((c + NBUF - 1) * BLOCK + tid);
            if (nidx >= VEC4) nidx = VEC4 - 1;          // clamp, keep issuing
            unsigned voff = nidx * 16u;
            unsigned dst  = lds_lane + (unsigned)((c + NBUF - 1) & (NBUF - 1)) * CHUNK_BYTES;
            asm volatile("global_load_async_to_lds_b128 %0, %1, %2 th:TH_LOAD_NT"
                         :: "v"(dst), "v"(voff), "s"(rowp) : "memory");
        }
        // Wait until chunk c has landed: outstanding newer chunks are
        // min(3, NCHUNK-1-c). Loads complete in order per wave.
        {
            int rem = NCHUNK - 1 - c;
            if (rem >= 3)      asm volatile("s_wait_asynccnt 0x3" ::: "memory");
            else if (rem == 2) asm volatile("s_wait_asynccnt 0x2" ::: "memory");
            else if (rem == 1) asm volatile("s_wait_asynccnt 0x1" ::: "memory");
            else               asm volatile("s_wait_asynccnt 0x0" ::: "memory");
        }

        int idx = c * BLOCK + tid;
        if (idx < VEC4) {
            float4 v = buf[c & (NBUF - 1)][tid];          // ds_load_b128
            float cmax = fmaxf(fmaxf(v.x, v.y), fmaxf(v.z, v.w));  // v_max3
            float nm  = fmaxf(m, cmax);
            float nmL = nm * LOG2E;
            // exp(x-nm) = exp2(fma(x, log2e, -nm*log2e)): 1 VALU + 1 TRANS each
            s = s * ex2(fmaf(m,   LOG2E, -nmL))
              +     ex2(fmaf(v.x, LOG2E, -nmL))
              +     ex2(fmaf(v.y, LOG2E, -nmL))
              +     ex2(fmaf(v.z, LOG2E, -nmL))
              +     ex2(fmaf(v.w, LOG2E, -nmL));
            m = nm;
        }
    }

    // Wave32 (m,s) reduction: merge running softmax states.
    for (int off = 16; off; off >>= 1) {
        float om = __shfl_xor(m, off, 32);
        float os = __shfl_xor(s, off, 32);
        float nm  = fmaxf(m, om);
        float nmL = nm * LOG2E;
        s = s * ex2(fmaf(m, LOG2E, -nmL)) + os * ex2(fmaf(om, LOG2E, -nmL));
        m = nm;
    }
    int wave = tid >> 5, lane = tid & 31;
    if (lane == 0) { wm[wave] = m; ws_[wave] = s; }
    __syncthreads();

    if (tid == 0) {
        float M = wm[0], S = ws_[0];
        #pragma unroll
        for (int w = 1; w < BLOCK / 32; ++w) {
            float nm  = fmaxf(M, wm[w]);
            float nmL = nm * LOG2E;
            S = S * ex2(fmaf(M, LOG2E, -nmL)) + ws_[w] * ex2(fmaf(wm[w], LOG2E, -nmL));
            M = nm;
        }
        // -log p(label) = log(sum exp(v - M)) + M - x_label
        row_loss[row] = lg2(S) * LN2 + M - xl;
    }
}

// ---------------------------------------------------------------------------
// Kernel 2: single workgroup, deterministic fixed-order reduction of the 8192
// per-row losses into the 17 outputs:
//   out[0]    = masked CE mean
//   out[1..8] = per-domain loss / (count*SEQ)  (0 for empty domains)
//   out[9..16]= samples per domain (as float)
// ---------------------------------------------------------------------------
__global__ __launch_bounds__(BLOCK) void ce_finalize_kernel(
    const float* __restrict__ row_loss,
    const int*   __restrict__ label_mask,    // bool -> int32 per harness convention
    const int*   __restrict__ domain_idxs,   // [BATCH], int32
    float*       __restrict__ out)
{
    const int tid = threadIdx.x;
    // Keep the 4 domain ids in scalar registers (no dynamic private indexing).
    const int d0 = domain_idxs[0], d1 = domain_idxs[1];
    const int d2 = domain_idxs[2], d3 = domain_idxs[3];

    float acc_ce = 0.0f, acc_cnt = 0.0f;
    float acc_dom[N_DOMAINS];
    #pragma unroll
    for (int k = 0; k < N_DOMAINS; ++k) acc_dom[k] = 0.0f;

    for (int r = tid; r < ROWS; r += BLOCK) {
        float L  = row_loss[r];
        float mk = (label_mask[r] != 0) ? 1.0f : 0.0f;
        acc_ce  += L * mk;
        acc_cnt += mk;
        int b = r >> 11;                       // r / SEQ
        int d = (b == 0) ? d0 : (b == 1) ? d1 : (b == 2) ? d2 : d3;
        #pragma unroll
        for (int k = 0; k < N_DOMAINS; ++k)    // branchless select keeps regs
            acc_dom[k] += (k == d) ? L : 0.0f;
    }

    // Wave32 sums, then 8-wave LDS combine (fixed order -> deterministic).
    for (int off = 16; off; off >>= 1) {
        acc_ce  += __shfl_xor(acc_ce, off, 32);
        acc_cnt += __shfl_xor(acc_cnt, off, 32);
        #pragma unroll
        for (int k = 0; k < N_DOMAINS; ++k)
            acc_dom[k] += __shfl_xor(acc_dom[k], off, 32);
    }
    __shared__ float red[BLOCK / 32][N_DOMAINS + 2];
    int wave = tid >> 5, lane = tid & 31;
    if (lane == 0) {
        red[wave][0] = acc_ce;
        red[wave][1] = acc_cnt;
        #pragma unroll
        for (int k = 0; k < N_DOMAINS; ++k) red[wave][2 + k] = acc_dom[k];
    }
    __syncthreads();

    if (tid == 0) {
        float ce = 0.0f, cnt = 0.0f, dsum[N_DOMAINS];
        #pragma unroll
        for (int k = 0; k < N_DOMAINS; ++k) dsum[k] = 0.0f;
        for (int w = 0; w < BLOCK / 32; ++w) {
            ce  += red[w][0];
            cnt += red[w][1];
            #pragma unroll
            for (int k = 0; k < N_DOMAINS; ++k) dsum[k] += red[w][2 + k];
        }
        out[0] = ce / cnt;
        #pragma unroll
        for (int k = 0; k < N_DOMAINS; ++k) {
            int count = (d0 == k) + (d1 == k) + (d2 == k) + (d3 == k);
            float denom = (float)(count * SEQ);
            out[1 + k] = (denom > 0.0f) ? dsum[k] / denom : 0.0f;
            out[9 + k] = (float)count;
        }
    }
}

// ---------------------------------------------------------------------------
// Inputs (setup_inputs order): 0=sharded_logits f32[4,2048,32000],
// 1=label_ids int[4,2048], 2=label_mask (bool->int) [4,2048],
// 3=domain_idxs int[4].  d_out: 17 floats.  d_ws: >= ROWS*4 bytes scratch.
// ---------------------------------------------------------------------------
extern "C" void kernel_launch(void* const* d_in, const int* in_sizes, int n_in,
                              void* d_out, int out_size, void* d_ws, size_t ws_size,
                              hipStream_t stream) {
    (void)in_sizes; (void)n_in; (void)out_size; (void)ws_size;
    const float* logits      = (const float*)d_in[0];
    const int*   labels      = (const int*)d_in[1];
    const int*   label_mask  = (const int*)d_in[2];
    const int*   domain_idxs = (const int*)d_in[3];
    float* row_loss = (float*)d_ws;               // ROWS floats (32 KB)

    ce_row_kernel<<<ROWS, BLOCK, 0, stream>>>(logits, labels, row_loss);
    ce_finalize_kernel<<<1, BLOCK, 0, stream>>>(row_loss, label_mask, domain_idxs,
                                                (float*)d_out);
}